// TimestepNorm_1503238554304
// MI455X (gfx1250) — compile-verified
//
#include <hip/hip_runtime.h>

typedef __attribute__((ext_vector_type(2))) float v2f;
typedef __attribute__((ext_vector_type(8))) float v8f;

#define NB   4
#define NL   4096
#define ND   2048
#define NG   16
#define NGS  128            // group size (ND / NG)
#define NCHK 64             // number of scan chunks
#define CHL  64             // chunk length (NCHK * CHL == NL)
#define EPSF   1e-5f
#define VFLOOR 1e-6f

#if __has_builtin(__builtin_amdgcn_wmma_f32_16x16x4_f32)
#define K1_WMMA 1
#endif

// ---------------------------------------------------------------------------
// K1: per-timestep group sums via V_WMMA_F32_16X16X4_F32.
//   A (16x4, f32): A[m][k] = x[row][m*128 + chunk*4 + k]  (m = group index)
//   B (4x16)     : all-ones  =>  D[m][n] accumulates group-m sum (all n equal)
// A-layout (ISA 7.12.2): lanes 0-15 hold K=0,1 in v0,v1; lanes 16-31 hold K=2,3.
// Result: lane 0 holds groups 0..7 in acc[0..7]; lane 16 holds groups 8..15.
// LDS shuffle -> coalesced transposed store gsumT[b][g][l].
// 8 loads kept in flight per wave (unroll 8) + WGP-scope prefetch ahead.
// ---------------------------------------------------------------------------
__global__ __launch_bounds__(128) void k1_group_sums(const float* __restrict__ x,
                                                     float* __restrict__ gsumT) {
    __shared__ float tile[4][NG];
#ifndef K1_WMMA
    __shared__ float part[4][32];
#endif
    const int w    = threadIdx.x >> 5;        // wave in block (row select)
    const int lane = threadIdx.x & 31;
    const int r    = blockIdx.x * 4 + w;      // row in [0, B*L)
    const int mrow  = lane & 15;
    const int khalf = lane >> 4;

    const float* p = x + (size_t)r * ND + mrow * NGS + khalf * 2;

#ifdef K1_WMMA
    v8f acc = {};
    v2f onesv; onesv.x = 1.0f; onesv.y = 1.0f;
    for (int co = 0; co < 4; ++co) {
        __builtin_prefetch(p + 128, 0, 3);    // 512B ahead, near-cache scope
        #pragma unroll
        for (int ci = 0; ci < 8; ++ci) {
            v2f a = *(const v2f*)p;
            // 8 args: (neg_a, A, neg_b, B, c_mod, C, reuse_a, reuse_b)
            acc = __builtin_amdgcn_wmma_f32_16x16x4_f32(false, a, false, onesv,
                                                        (short)0, acc, false, false);
            p += 4;
        }
    }
    if (lane == 0) {
        #pragma unroll
        for (int i = 0; i < 8; ++i) tile[w][i] = acc[i];
    } else if (lane == 16) {
        #pragma unroll
        for (int i = 0; i < 8; ++i) tile[w][8 + i] = acc[i];
    }
    __syncthreads();
    if (threadIdx.x < 64) {
        const int g  = threadIdx.x >> 2;
        const int dl = threadIdx.x & 3;
        const int r0 = blockIdx.x * 4;
        const int b  = r0 / NL;
        const int l  = (r0 % NL) + dl;
        gsumT[((size_t)(b * NG + g)) * NL + l] = tile[dl][g];
    }
#else
    float partial = 0.0f;
    #pragma unroll 8
    for (int c = 0; c < 32; ++c) {
        v2f a = *(const v2f*)p;
        partial += a.x + a.y;
        p += 4;
    }
    part[w][lane] = partial;
    __syncthreads();
    if (threadIdx.x < 64) {
        const int g  = threadIdx.x >> 2;
        const int dl = threadIdx.x & 3;
        const int r0 = blockIdx.x * 4;
        const int b  = r0 / NL;
        const int l  = (r0 % NL) + dl;
        gsumT[((size_t)(b * NG + g)) * NL + l] = part[dl][g] + part[dl][g + 16];
    }
    (void)tile;
#endif
}

// ---------------------------------------------------------------------------
// K2: chunked (two-level) Welford prefix scan. Single block, 5 phases.
// The reference's cumsum-based formulation is prefix-sum decomposable:
//   mean_l from prefix(gmean*valid), prefix(valid);
//   term_l = (g_l - mean_{l-1})(g_l - mean_l)*valid; m2_l from prefix(term).
// All scan state (~4 MB) is L2-resident (192 MB L2).
// ---------------------------------------------------------------------------
__global__ __launch_bounds__(1024) void k2_scan(
    const float* __restrict__ gsumT, const int* __restrict__ prev_count,
    const float* __restrict__ prev_mean, const float* __restrict__ prev_var,
    const unsigned char* __restrict__ mask,
    float* __restrict__ meanArr, float* __restrict__ rstdArr, float* __restrict__ termArr,
    float* __restrict__ sumAgg, float* __restrict__ termAgg, float* __restrict__ cntAgg,
    float* __restrict__ sumOff, float* __restrict__ termOff, float* __restrict__ cntOff,
    int* __restrict__ outCnt, float* __restrict__ outMean, float* __restrict__ outVar) {
    const int tid = threadIdx.x;
    const float invgs = 1.0f / (float)NGS;

    // Phase 1: chunk-level reductions (parallel over B*G*NCHK = 4096 items)
    for (int item = tid; item < NB * NG * NCHK; item += 1024) {
        const int c = item % NCHK, bg = item / NCHK, g = bg % NG, b = bg / NG;
        const float* gp = gsumT + (size_t)bg * NL + c * CHL;
        const unsigned char* mp = mask + b * NL + c * CHL;
        float s = 0.0f, cnt = 0.0f;
        for (int i = 0; i < CHL; ++i) {
            const float v = mp[i] ? 1.0f : 0.0f;
            s += gp[i] * invgs * v;
            cnt += v;
        }
        sumAgg[item] = s;
        if (g == 0) cntAgg[b * NCHK + c] = cnt;
    }
    __threadfence(); __syncthreads();

    // Phase 2: serial exclusive prefix over chunks (64 threads; counts on 4 more)
    if (tid < NB * NG) {
        const int b = tid / NG;
        const float pc = (float)prev_count[b];
        float off = prev_mean[tid] * pc;                     // prev_sum
        for (int c = 0; c < NCHK; ++c) { sumOff[tid * NCHK + c] = off; off += sumAgg[tid * NCHK + c]; }
    } else if (tid >= 128 && tid < 128 + NB) {
        const int b = tid - 128;
        float off = (float)prev_count[b];
        for (int c = 0; c < NCHK; ++c) { cntOff[b * NCHK + c] = off; off += cntAgg[b * NCHK + c]; }
        int itot = 0;
        for (int l = 0; l < NL; ++l) itot += (mask[b * NL + l] ? 1 : 0);
        outCnt[b] = prev_count[b] + itot;
    }
    __threadfence(); __syncthreads();

    // Phase 3: within-chunk mean pass + Welford term computation
    for (int item = tid; item < NB * NG * NCHK; item += 1024) {
        const int c = item % NCHK, bg = item / NCHK, g = bg % NG, b = bg / NG;
        const int l0 = c * CHL;
        const float pm = prev_mean[bg];
        float cnt = cntOff[b * NCHK + c];
        float sum = sumOff[item];
        float meanprev = (cnt > 0.0f) ? sum / fmaxf(cnt, 1.0f) : pm;  // mean_{l0-1}
        float ts = 0.0f;
        for (int i = 0; i < CHL; ++i) {
            const int l = l0 + i;
            const float v  = mask[b * NL + l] ? 1.0f : 0.0f;
            const float gm = gsumT[(size_t)bg * NL + l] * invgs;
            sum += gm * v; cnt += v;
            const float mean = (cnt > 0.0f) ? sum / fmaxf(cnt, 1.0f) : pm;
            const float term = (gm - meanprev) * (gm - mean) * v;
            ts += term;
            const size_t o = (size_t)(b * NL + l) * NG + g;
            meanArr[o] = mean;
            termArr[o] = term;
            meanprev = mean;
        }
        termAgg[item] = ts;
    }
    __threadfence(); __syncthreads();

    // Phase 4: serial exclusive prefix of M2 terms
    if (tid < NB * NG) {
        const int b = tid / NG;
        const float pc = (float)prev_count[b];
        float off = prev_var[tid] * fmaxf(pc, 1.0f);          // prev_m2
        for (int c = 0; c < NCHK; ++c) { termOff[tid * NCHK + c] = off; off += termAgg[tid * NCHK + c]; }
    }
    __threadfence(); __syncthreads();

    // Phase 5: variance / rstd + final stats outputs
    for (int item = tid; item < NB * NG * NCHK; item += 1024) {
        const int c = item % NCHK, bg = item / NCHK, g = bg % NG, b = bg / NG;
        const int l0 = c * CHL;
        const float pv = prev_var[bg];
        float cnt = cntOff[b * NCHK + c];
        float m2  = termOff[item];
        for (int i = 0; i < CHL; ++i) {
            const int l = l0 + i;
            const float v = mask[b * NL + l] ? 1.0f : 0.0f;
            cnt += v;
            const size_t o = (size_t)(b * NL + l) * NG + g;
            m2 += termArr[o];
            float var = (cnt > 0.0f) ? m2 / fmaxf(cnt, 1.0f) : pv;
            var = fmaxf(var, VFLOOR);
            rstdArr[o] = rsqrtf(var + EPSF);
            if (l == NL - 1) { outMean[bg] = meanArr[o]; outVar[bg] = var; }
        }
    }
}

// ---------------------------------------------------------------------------
// K3: normalize + affine, float4-vectorized (stream x -> y at full HBM BW).
// ---------------------------------------------------------------------------
__global__ __launch_bounds__(256) void k3_normalize(
    const float* __restrict__ x, const float* __restrict__ meanArr,
    const float* __restrict__ rstdArr, const float* __restrict__ wgt,
    const float* __restrict__ bias, float* __restrict__ y) {
    const size_t i4 = (size_t)blockIdx.x * 256 + threadIdx.x;
    const size_t e  = i4 * 4;
    const int    d  = (int)(e & (ND - 1));
    const size_t r  = e >> 11;                 // row = e / ND
    const int    g  = d >> 7;                  // group = d / 128
    const float  m  = meanArr[r * NG + g];
    const float  rs = rstdArr[r * NG + g];
    const float4 xv = *(const float4*)(x + e);
    const float4 wv = *(const float4*)(wgt + d);
    const float4 bv = *(const float4*)(bias + d);
    float4 o;
    o.x = (xv.x - m) * rs * (wv.x + 1.0f) + bv.x;
    o.y = (xv.y - m) * rs * (wv.y + 1.0f) + bv.y;
    o.z = (xv.z - m) * rs * (wv.z + 1.0f) + bv.z;
    o.w = (xv.w - m) * rs * (wv.w + 1.0f) + bv.w;
    *(float4*)(y + e) = o;
}

// ---------------------------------------------------------------------------
extern "C" void kernel_launch(void* const* d_in, const int* in_sizes, int n_in,
                              void* d_out, int out_size, void* d_ws, size_t ws_size,
                              hipStream_t stream) {
    const float*         x          = (const float*)d_in[0];
    const int*           prev_count = (const int*)d_in[1];
    const float*         prev_mean  = (const float*)d_in[2];
    const float*         prev_var   = (const float*)d_in[3];
    const unsigned char* mask       = (const unsigned char*)d_in[4];
    const float*         weight     = (const float*)d_in[5];
    const float*         bias       = (const float*)d_in[6];

    // workspace layout (floats)
    float* ws      = (float*)d_ws;
    float* gsumT   = ws;                               // B*G*L   = 262144
    float* meanArr = gsumT   + (size_t)NB * NG * NL;   // B*L*G   = 262144
    float* rstdArr = meanArr + (size_t)NB * NL * NG;   // B*L*G
    float* termArr = rstdArr + (size_t)NB * NL * NG;   // B*L*G
    float* sumAgg  = termArr + (size_t)NB * NL * NG;   // B*G*NCHK = 4096
    float* termAgg = sumAgg  + NB * NG * NCHK;
    float* cntAgg  = termAgg + NB * NG * NCHK;         // B*NCHK   = 256
    float* sumOff  = cntAgg  + NB * NCHK;
    float* termOff = sumOff  + NB * NG * NCHK;
    float* cntOff  = termOff + NB * NG * NCHK;

    // output layout: y | new_count(int) | final_mean | final_var
    float* y       = (float*)d_out;
    int*   outCnt  = (int*)(y + (size_t)NB * NL * ND);
    float* outMean = (float*)(outCnt + NB);
    float* outVar  = outMean + NB * NG;

    // K1: group sums (WMMA f32 16x16x4 reduction), 4 rows per block
    k1_group_sums<<<(NB * NL) / 4, 128, 0, stream>>>(x, gsumT);

    // K2: chunked Welford prefix scan (single block, L2-resident state)
    k2_scan<<<1, 1024, 0, stream>>>(gsumT, prev_count, prev_mean, prev_var, mask,
                                    meanArr, rstdArr, termArr,
                                    sumAgg, termAgg, cntAgg,
                                    sumOff, termOff, cntOff,
                                    outCnt, outMean, outVar);

    // K3: normalize + affine
    const size_t total4 = (size_t)NB * NL * ND / 4;    // 8388608
    k3_normalize<<<(unsigned)(total4 / 256), 256, 0, stream>>>(x, meanArr, rstdArr,
                                                               weight, bias, y);
}